// GatedGCNGraphGymLayer_60430189855387
// MI455X (gfx1250) — compile-verified
//
#include <hip/hip_runtime.h>

#define DD      128
#define NNODES  50000
#define NEDGES  800000
#define LSTRIDE 132            // 128 + 4 dword pad -> conflict-free A-fragment reads
#define BN_EPS  1e-5f
#define AGG_EPS 1e-6f

typedef __attribute__((ext_vector_type(2))) float v2f;
typedef __attribute__((ext_vector_type(8))) float v8f;

// ---------------------------------------------------------------------------
// Async cache->LDS copy of a 16x128 fp32 tile (row-padded to LSTRIDE dwords).
// Uses gfx1250 GLOBAL_LOAD_ASYNC_TO_LDS_B128 (ASYNCcnt-tracked, no VGPR data).
// LDS VGPR operand = low 32 bits of the generic shared-memory pointer
// (generic LDS addresses map to LDS offsets by dropping the aperture bits).
// ---------------------------------------------------------------------------
__device__ __forceinline__ void async_tile_load_16x128(
    float* lds_tile, const float* gsrc, int tid)
{
    #pragma unroll
    for (int i = tid; i < 16 * DD / 4; i += 256) {
        const int r = (i * 4) / DD;
        const int c = (i * 4) % DD;
        const unsigned lofs = (unsigned)(size_t)(&lds_tile[r * LSTRIDE + c]);
        const unsigned long long ga = (unsigned long long)(gsrc + (size_t)i * 4);
        asm volatile("global_load_async_to_lds_b128 %0, %1, off"
                     :: "v"(lofs), "v"(ga) : "memory");
    }
    asm volatile("s_wait_asynccnt 0x0" ::: "memory");
}

// ---------------------------------------------------------------------------
// Node GEMMs: Ax,Bx,Dx,Ex = x @ {WA,WB,WD,WE} + bias   (shared 16x128 A tile)
// ---------------------------------------------------------------------------
__global__ __launch_bounds__(256) void node_gemm_kernel(
    const float* __restrict__ x,
    const float* __restrict__ WA, const float* __restrict__ bA,
    const float* __restrict__ WB, const float* __restrict__ bB,
    const float* __restrict__ WD, const float* __restrict__ bD,
    const float* __restrict__ WE, const float* __restrict__ bE,
    float* __restrict__ Ax, float* __restrict__ Bx,
    float* __restrict__ Dx, float* __restrict__ Ex)
{
    __shared__ float lx[16 * LSTRIDE];
    const int row0 = blockIdx.x * 16;
    const int tid  = threadIdx.x;

    async_tile_load_16x128(lx, x + (size_t)row0 * DD, tid);
    __syncthreads();

    const int wave = tid >> 5;
    const int lane = tid & 31;
    const int hb   = lane >> 4;     // half-wave (K offset 0 or 2)
    const int ln   = lane & 15;
    const int n0   = wave * 16;

    const float* Ws[4]   = {WA, WB, WD, WE};
    const float* bs[4]   = {bA, bB, bD, bE};
    float*       outs[4] = {Ax, Bx, Dx, Ex};

    #pragma unroll
    for (int m = 0; m < 4; ++m) {
        const float* __restrict__ W = Ws[m];
        const float bias = bs[m][n0 + ln];
        v8f acc;
        #pragma unroll
        for (int i = 0; i < 8; ++i) acc[i] = bias;

        #pragma unroll
        for (int k = 0; k < DD; k += 4) {
            const int kk = k + hb * 2;
            v2f a, b;
            a.x = lx[ln * LSTRIDE + kk];
            a.y = lx[ln * LSTRIDE + kk + 1];
            b.x = W[(size_t)kk * DD + n0 + ln];
            b.y = W[(size_t)(kk + 1) * DD + n0 + ln];
            acc = __builtin_amdgcn_wmma_f32_16x16x4_f32(
                false, a, false, b, (short)0, acc, false, false);
        }
        float* __restrict__ out = outs[m];
        #pragma unroll
        for (int i = 0; i < 8; ++i) {
            const int r = row0 + i + hb * 8;
            out[(size_t)r * DD + n0 + ln] = acc[i];
        }
    }
}

// ---------------------------------------------------------------------------
// Edge GEMM + message: e_ij = Dx[dst] + Ex[src] + (e @ WC + bC)
//   writes e_ij into d_out's e region, scatters sigma & sigma*Bx[src].
// ---------------------------------------------------------------------------
__global__ __launch_bounds__(256) void edge_msg_kernel(
    const float* __restrict__ e,
    const int*   __restrict__ src_idx, const int* __restrict__ dst_idx,
    const float* __restrict__ WC, const float* __restrict__ bC,
    const float* __restrict__ Bx, const float* __restrict__ Dxm,
    const float* __restrict__ Exm,
    float* __restrict__ e_ij_out,
    float* __restrict__ num, float* __restrict__ den)
{
    __shared__ float le[16 * LSTRIDE];
    __shared__ int ssrc[16], sdst[16];
    const int row0 = blockIdx.x * 16;
    const int tid  = threadIdx.x;

    async_tile_load_16x128(le, e + (size_t)row0 * DD, tid);
    if (tid < 16) {
        ssrc[tid] = src_idx[row0 + tid];
        sdst[tid] = dst_idx[row0 + tid];
    }
    __syncthreads();

    const int wave = tid >> 5;
    const int lane = tid & 31;
    const int hb   = lane >> 4;
    const int ln   = lane & 15;
    const int n0   = wave * 16;

    const float bias = bC[n0 + ln];
    v8f acc;
    #pragma unroll
    for (int i = 0; i < 8; ++i) acc[i] = bias;

    #pragma unroll
    for (int k = 0; k < DD; k += 4) {
        const int kk = k + hb * 2;
        v2f a, b;
        a.x = le[ln * LSTRIDE + kk];
        a.y = le[ln * LSTRIDE + kk + 1];
        b.x = WC[(size_t)kk * DD + n0 + ln];
        b.y = WC[(size_t)(kk + 1) * DD + n0 + ln];
        acc = __builtin_amdgcn_wmma_f32_16x16x4_f32(
            false, a, false, b, (short)0, acc, false, false);
    }

    const int col = n0 + ln;
    #pragma unroll
    for (int i = 0; i < 8; ++i) {
        const int rloc = i + hb * 8;
        const int s = ssrc[rloc];
        const int d = sdst[rloc];
        const float v = acc[i] + Dxm[(size_t)d * DD + col]
                               + Exm[(size_t)s * DD + col];
        e_ij_out[(size_t)(row0 + rloc) * DD + col] = v;
        const float sg = 1.0f / (1.0f + __expf(-v));
        atomicAdd(&den[(size_t)d * DD + col], sg);
        atomicAdd(&num[(size_t)d * DD + col], sg * Bx[(size_t)s * DD + col]);
    }
}

// ---------------------------------------------------------------------------
// h = Ax + num / (den + eps)   (written into d_out's x region), float4
// ---------------------------------------------------------------------------
__global__ __launch_bounds__(256) void aggr_kernel(
    const float4* __restrict__ Ax, const float4* __restrict__ num,
    const float4* __restrict__ den, float4* __restrict__ h)
{
    const size_t i = (size_t)blockIdx.x * 256 + threadIdx.x;
    if (i >= (size_t)NNODES * DD / 4) return;
    const float4 a = Ax[i];
    const float4 n = num[i];
    const float4 d = den[i];
    float4 o;
    o.x = a.x + n.x / (d.x + AGG_EPS);
    o.y = a.y + n.y / (d.y + AGG_EPS);
    o.z = a.z + n.z / (d.z + AGG_EPS);
    o.w = a.w + n.w / (d.w + AGG_EPS);
    h[i] = o;
}

// ---------------------------------------------------------------------------
// Per-feature (column) sum / sumsq: 32 lanes x float4 = one coalesced row,
// 8 row-streams per block, LDS tree, then per-feature global atomics.
// ---------------------------------------------------------------------------
__global__ __launch_bounds__(256) void colstats_kernel(
    const float* __restrict__ v, long n_rows,
    float* __restrict__ sum, float* __restrict__ sumsq)
{
    const int f4  = (threadIdx.x & 31) * 4;
    const int sub = threadIdx.x >> 5;   // 0..7
    float4 s = {0.f, 0.f, 0.f, 0.f};
    float4 q = {0.f, 0.f, 0.f, 0.f};
    for (long r = (long)blockIdx.x * 8 + sub; r < n_rows;
         r += (long)gridDim.x * 8) {
        const float4 x = *(const float4*)(v + (size_t)r * DD + f4);
        s.x += x.x; s.y += x.y; s.z += x.z; s.w += x.w;
        q.x += x.x * x.x; q.y += x.y * x.y;
        q.z += x.z * x.z; q.w += x.w * x.w;
    }
    __shared__ float4 ls[256], lq[256];
    ls[threadIdx.x] = s;
    lq[threadIdx.x] = q;
    __syncthreads();
    if (sub == 0) {
        const int t = threadIdx.x;   // 0..31
        float4 S = ls[t], Q = lq[t];
        #pragma unroll
        for (int j = 1; j < 8; ++j) {
            const float4 a = ls[j * 32 + t];
            const float4 b = lq[j * 32 + t];
            S.x += a.x; S.y += a.y; S.z += a.z; S.w += a.w;
            Q.x += b.x; Q.y += b.y; Q.z += b.z; Q.w += b.w;
        }
        atomicAdd(&sum[f4 + 0], S.x); atomicAdd(&sum[f4 + 1], S.y);
        atomicAdd(&sum[f4 + 2], S.z); atomicAdd(&sum[f4 + 3], S.w);
        atomicAdd(&sumsq[f4 + 0], Q.x); atomicAdd(&sumsq[f4 + 1], Q.y);
        atomicAdd(&sumsq[f4 + 2], Q.z); atomicAdd(&sumsq[f4 + 3], Q.w);
    }
}

// ---------------------------------------------------------------------------
// In-place BatchNorm (batch stats) + ReLU, float4
// ---------------------------------------------------------------------------
__global__ __launch_bounds__(256) void bn_relu_kernel(
    float* __restrict__ v, long n_rows,
    const float* __restrict__ sum, const float* __restrict__ sumsq,
    const float* __restrict__ gamma, const float* __restrict__ beta)
{
    const size_t i4 = (size_t)blockIdx.x * 256 + threadIdx.x;
    const size_t total4 = (size_t)n_rows * DD / 4;
    if (i4 >= total4) return;
    const int f0 = (int)((i4 * 4) & 127);
    const float inv_n = 1.0f / (float)n_rows;
    const float4 s = *(const float4*)(sum + f0);
    const float4 q = *(const float4*)(sumsq + f0);
    const float4 g = *(const float4*)(gamma + f0);
    const float4 b = *(const float4*)(beta + f0);
    float4 val = ((const float4*)v)[i4];

    float m, var, inv, y;
    m = s.x * inv_n; var = q.x * inv_n - m * m; inv = rsqrtf(var + BN_EPS);
    y = g.x * (val.x - m) * inv + b.x; val.x = y > 0.f ? y : 0.f;
    m = s.y * inv_n; var = q.y * inv_n - m * m; inv = rsqrtf(var + BN_EPS);
    y = g.y * (val.y - m) * inv + b.y; val.y = y > 0.f ? y : 0.f;
    m = s.z * inv_n; var = q.z * inv_n - m * m; inv = rsqrtf(var + BN_EPS);
    y = g.z * (val.z - m) * inv + b.z; val.z = y > 0.f ? y : 0.f;
    m = s.w * inv_n; var = q.w * inv_n - m * m; inv = rsqrtf(var + BN_EPS);
    y = g.w * (val.w - m) * inv + b.w; val.w = y > 0.f ? y : 0.f;

    ((float4*)v)[i4] = val;
}

// ---------------------------------------------------------------------------
extern "C" void kernel_launch(void* const* d_in, const int* in_sizes, int n_in,
                              void* d_out, int out_size, void* d_ws, size_t ws_size,
                              hipStream_t stream)
{
    const float* x          = (const float*)d_in[0];
    const float* e          = (const float*)d_in[1];
    const int*   edge_index = (const int*)  d_in[2];
    const float* WA = (const float*)d_in[3];  const float* bA = (const float*)d_in[4];
    const float* WB = (const float*)d_in[5];  const float* bB = (const float*)d_in[6];
    const float* WC = (const float*)d_in[7];  const float* bC = (const float*)d_in[8];
    const float* WD = (const float*)d_in[9];  const float* bD = (const float*)d_in[10];
    const float* WE = (const float*)d_in[11]; const float* bE = (const float*)d_in[12];
    const float* gamma_x = (const float*)d_in[13];
    const float* beta_x  = (const float*)d_in[14];
    const float* gamma_e = (const float*)d_in[15];
    const float* beta_e  = (const float*)d_in[16];

    const int* src_idx = edge_index;            // edge_index[0]
    const int* dst_idx = edge_index + NEDGES;   // edge_index[1]

    float* xout = (float*)d_out;                      // [NNODES, 128]
    float* eout = xout + (size_t)NNODES * DD;         // [NEDGES, 128]

    const size_t NM = (size_t)NNODES * DD;
    const size_t EM = (size_t)NEDGES * DD;
    float* ws    = (float*)d_ws;
    float* Ax    = ws;
    float* Bx    = ws + 1 * NM;
    float* Dx    = ws + 2 * NM;
    float* Ex    = ws + 3 * NM;
    float* num   = ws + 4 * NM;
    float* den   = ws + 5 * NM;
    float* stats = ws + 6 * NM;   // [sum_x 128 | sumsq_x 128 | sum_e 128 | sumsq_e 128]

    // zero num, den and stats (contiguous region)
    hipMemsetAsync(num, 0, (2 * NM + 512) * sizeof(float), stream);

    // 1) node GEMMs (50000 rows / 16 per block = 3125 blocks)
    node_gemm_kernel<<<NNODES / 16, 256, 0, stream>>>(
        x, WA, bA, WB, bB, WD, bD, WE, bE, Ax, Bx, Dx, Ex);

    // 2) edge GEMM + message + scatter (800000 / 16 = 50000 blocks)
    edge_msg_kernel<<<NEDGES / 16, 256, 0, stream>>>(
        e, src_idx, dst_idx, WC, bC, Bx, Dx, Ex, eout, num, den);

    // 3) h = Ax + aggr  -> x region of d_out
    aggr_kernel<<<(int)((NM / 4 + 255) / 256), 256, 0, stream>>>(
        (const float4*)Ax, (const float4*)num, (const float4*)den, (float4*)xout);

    // 4) BN statistics (column sums)
    colstats_kernel<<<256, 256, 0, stream>>>(xout, (long)NNODES, stats + 0,   stats + 128);
    colstats_kernel<<<2048, 256, 0, stream>>>(eout, (long)NEDGES, stats + 256, stats + 384);

    // 5) BN + ReLU in place
    bn_relu_kernel<<<(int)((NM / 4 + 255) / 256), 256, 0, stream>>>(
        xout, (long)NNODES, stats + 0, stats + 128, gamma_x, beta_x);
    bn_relu_kernel<<<(int)((EM / 4 + 255) / 256), 256, 0, stream>>>(
        eout, (long)NEDGES, stats + 256, stats + 384, gamma_e, beta_e);
}